// GPR_52630529245707
// MI455X (gfx1250) — compile-verified
//
#include <hip/hip_runtime.h>
#include <hip/hip_bf16.h>

typedef float v2f __attribute__((ext_vector_type(2)));
typedef float v8f __attribute__((ext_vector_type(8)));

#define NN 8192
#define BB 4096

static __device__ __forceinline__ v8f wmma4(v2f a, v2f b, v8f c) {
  // D = A(16x4) * B(4x16) + C, fp32
  return __builtin_amdgcn_wmma_f32_16x16x4_f32(false, a, false, b, (short)0, c, false, false);
}

static __device__ __forceinline__ float lrelu(float x) {
  return x >= 0.f ? x : 0.01f * x;
}

// ---------------- zero fill ----------------
__global__ void k_zero(float* __restrict__ p, long n) {
  long i = (long)blockIdx.x * blockDim.x + threadIdx.x;
  long stride = (long)gridDim.x * blockDim.x;
  for (; i < n; i += stride) p[i] = 0.f;
}

// ---------------- Wp/Wq projections: Wpq[n, k*64+d] = p@Wg[k]^T + bg[k] ----------------
__global__ void k_wpq(const float* __restrict__ pe, const float* __restrict__ qe,
                      const float* __restrict__ Wg, const float* __restrict__ bg,
                      float* __restrict__ Wp, float* __restrict__ Wq) {
  int idx = blockIdx.x * 256 + threadIdx.x;       // 2 * 8192 * 128
  int which = idx >> 20;                          // 8192*128 = 1<<20
  int rem = idx & ((1 << 20) - 1);
  int n = rem >> 7, c = rem & 127;                // c = k*64 + d
  const float* src = which ? qe : pe;
  const float* w = Wg + (long)c * 64;             // Wg[(k*64+d)*64 + t]
  const float* row = src + (long)n * 64;
  float acc = bg[c];
  #pragma unroll
  for (int t = 0; t < 64; t++) acc += row[t] * w[t];
  (which ? Wq : Wp)[(long)n * 128 + c] = acc;
}

// ---------------- P aggregation: Pagg += e^T @ Wp ; Dout += colsum(e) ----------------
__global__ void __launch_bounds__(32) k_agg_p(const float* __restrict__ e,
                                              const float* __restrict__ Wp,
                                              float* __restrict__ Pagg,
                                              float* __restrict__ Dout) {
  const int lane = threadIdx.x;
  const int h = lane >> 4, lm = lane & 15;
  const int jbase = blockIdx.x * 32;
  const int i0s = blockIdx.y * (NN / 8);
  const int i0e = i0s + (NN / 8);

  v8f c[2][8];
  #pragma unroll
  for (int jf = 0; jf < 2; jf++)
    #pragma unroll
    for (int t = 0; t < 8; t++)
      c[jf][t] = (v8f){0.f,0.f,0.f,0.f,0.f,0.f,0.f,0.f};
  float sA[2] = {0.f, 0.f};

  for (int i0 = i0s; i0 < i0e; i0 += 4) {
    v2f a[2];
    #pragma unroll
    for (int jf = 0; jf < 2; jf++) {
      int col = jbase + 16 * jf + lm;
      float a0 = e[(long)(i0 + 2 * h) * NN + col];
      float a1 = e[(long)(i0 + 2 * h + 1) * NN + col];
      a[jf][0] = a0; a[jf][1] = a1;
      sA[jf] += a0 + a1;
    }
    v2f b[8];
    #pragma unroll
    for (int t = 0; t < 8; t++) {
      int dc = 16 * t + lm;
      b[t][0] = Wp[(long)(i0 + 2 * h) * 128 + dc];
      b[t][1] = Wp[(long)(i0 + 2 * h + 1) * 128 + dc];
    }
    #pragma unroll
    for (int jf = 0; jf < 2; jf++)
      #pragma unroll
      for (int t = 0; t < 8; t++)
        c[jf][t] = wmma4(a[jf], b[t], c[jf][t]);
  }

  #pragma unroll
  for (int jf = 0; jf < 2; jf++)
    unsafeAtomicAdd(&Dout[jbase + 16 * jf + lm], sA[jf]);

  #pragma unroll
  for (int jf = 0; jf < 2; jf++)
    #pragma unroll
    for (int t = 0; t < 8; t++)
      #pragma unroll
      for (int v = 0; v < 8; v++) {
        int row = jbase + 16 * jf + v + 8 * h;
        int col = 16 * t + lm;
        unsafeAtomicAdd(&Pagg[(long)row * 128 + col], c[jf][t][v]);
      }
}

// ---------------- Q aggregation: Qagg += e @ Wq ; Din += rowsum(e) ----------------
__global__ void __launch_bounds__(32) k_agg_q(const float* __restrict__ e,
                                              const float* __restrict__ Wq,
                                              float* __restrict__ Qagg,
                                              float* __restrict__ Din) {
  const int lane = threadIdx.x;
  const int h = lane >> 4, lm = lane & 15;
  const int jbase = blockIdx.x * 32;
  const int i0s = blockIdx.y * (NN / 8);
  const int i0e = i0s + (NN / 8);

  v8f c[2][8];
  #pragma unroll
  for (int jf = 0; jf < 2; jf++)
    #pragma unroll
    for (int t = 0; t < 8; t++)
      c[jf][t] = (v8f){0.f,0.f,0.f,0.f,0.f,0.f,0.f,0.f};
  float sA[2] = {0.f, 0.f};

  for (int i0 = i0s; i0 < i0e; i0 += 4) {
    v2f a[2];
    #pragma unroll
    for (int jf = 0; jf < 2; jf++) {
      int row = jbase + 16 * jf + lm;
      float2 av = *reinterpret_cast<const float2*>(&e[(long)row * NN + i0 + 2 * h]);
      a[jf][0] = av.x; a[jf][1] = av.y;
      sA[jf] += av.x + av.y;
    }
    v2f b[8];
    #pragma unroll
    for (int t = 0; t < 8; t++) {
      int dc = 16 * t + lm;
      b[t][0] = Wq[(long)(i0 + 2 * h) * 128 + dc];
      b[t][1] = Wq[(long)(i0 + 2 * h + 1) * 128 + dc];
    }
    #pragma unroll
    for (int jf = 0; jf < 2; jf++)
      #pragma unroll
      for (int t = 0; t < 8; t++)
        c[jf][t] = wmma4(a[jf], b[t], c[jf][t]);
  }

  #pragma unroll
  for (int jf = 0; jf < 2; jf++)
    unsafeAtomicAdd(&Din[jbase + 16 * jf + lm], sA[jf]);

  #pragma unroll
  for (int jf = 0; jf < 2; jf++)
    #pragma unroll
    for (int t = 0; t < 8; t++)
      #pragma unroll
      for (int v = 0; v < 8; v++) {
        int row = jbase + 16 * jf + v + 8 * h;
        int col = 16 * t + lm;
        unsafeAtomicAdd(&Qagg[(long)row * 128 + col], c[jf][t][v]);
      }
}

// ---------------- epilogue: new = leaky_relu(agg / deg) in place ----------------
__global__ void k_epi(float* __restrict__ P, float* __restrict__ Q,
                      const float* __restrict__ Dout, const float* __restrict__ Din) {
  long idx = (long)blockIdx.x * 256 + threadIdx.x;  // N*128
  int n = (int)(idx >> 7);
  P[idx] = lrelu(P[idx] / Dout[n]);
  Q[idx] = lrelu(Q[idx] / Din[n]);
}

// ---------------- H = new_p1 @ Wdec^T + bdec ----------------
__global__ void k_h(const float* __restrict__ P, const float* __restrict__ Wdec,
                    const float* __restrict__ bdec, float* __restrict__ H) {
  long idx = (long)blockIdx.x * 256 + threadIdx.x;  // N*64
  int n = (int)(idx >> 6), d = (int)(idx & 63);
  const float* row = P + (long)n * 128 + 64;        // p_k[-1]
  const float* w = Wdec + (long)d * 64;
  float acc = bdec[d];
  #pragma unroll
  for (int t = 0; t < 64; t++) acc += row[t] * w[t];
  H[idx] = acc;
}

// ---------------- e_hat = (H @ q1^T) * a*d^b*exp(c*d) ----------------
__global__ void __launch_bounds__(32) k_ehat(const float* __restrict__ H,
                                             const float* __restrict__ Q,
                                             const float* __restrict__ dist,
                                             const float* __restrict__ pa,
                                             const float* __restrict__ pb,
                                             const float* __restrict__ pc,
                                             float* __restrict__ out) {
  const int lane = threadIdx.x, h = lane >> 4, lm = lane & 15;
  const int m0 = blockIdx.x * 16, n0 = blockIdx.y * 16;
  v8f c = (v8f){0.f,0.f,0.f,0.f,0.f,0.f,0.f,0.f};
  #pragma unroll
  for (int s = 0; s < 16; s++) {
    int k0 = s * 4 + 2 * h;
    float2 av = *reinterpret_cast<const float2*>(&H[(long)(m0 + lm) * 64 + k0]);
    float2 bv = *reinterpret_cast<const float2*>(&Q[(long)(n0 + lm) * 128 + 64 + k0]);
    v2f a, b; a[0] = av.x; a[1] = av.y; b[0] = bv.x; b[1] = bv.y;
    c = wmma4(a, b, c);
  }
  float fa = pa[0], fb = pb[0], fc = pc[0];
  #pragma unroll
  for (int v = 0; v < 8; v++) {
    long row = m0 + v + 8 * h;
    long col = n0 + lm;
    float d = dist[row * NN + col];
    float fx = fa * __expf(fb * __logf(d) + fc * d);
    out[row * NN + col] = c[v] * fx;
  }
}

// ---------------- column sums of new_p (both k) ----------------
__global__ void k_colsum(const float* __restrict__ P, float* __restrict__ cs) {
  __shared__ float red[256];
  int c = blockIdx.x;  // 0..127
  float acc = 0.f;
  for (int n = threadIdx.x; n < NN; n += 256) acc += P[(long)n * 128 + c];
  red[threadIdx.x] = acc;
  __syncthreads();
  for (int s = 128; s > 0; s >>= 1) {
    if (threadIdx.x < s) red[threadIdx.x] += red[threadIdx.x + s];
    __syncthreads();
  }
  if (threadIdx.x == 0) cs[c] = red[0];
}

// ---------------- s[i*64+d] = colsumP_i @ Wo[i]^T + N*bo[i] ----------------
__global__ void k_svec(const float* __restrict__ cs, const float* __restrict__ Wo,
                       const float* __restrict__ bo, float* __restrict__ s) {
  int tid = threadIdx.x;  // 128
  int i = tid >> 6, d = tid & 63;
  float acc = (float)NN * bo[tid];
  const float* w = Wo + (long)i * 4096 + (long)d * 64;
  const float* r = cs + i * 64;
  #pragma unroll
  for (int t = 0; t < 64; t++) acc += r[t] * w[t];
  s[tid] = acc;
}

// ---------------- user tower layer 0 / layer 1 ----------------
__global__ void k_u0(const int* __restrict__ users, const float* __restrict__ ue,
                     const float* __restrict__ Wu, const float* __restrict__ s,
                     float* __restrict__ U) {
  long idx = (long)blockIdx.x * 256 + threadIdx.x;  // B*64
  int b = (int)(idx >> 6), d = (int)(idx & 63);
  const float* row = ue + (long)users[b] * 64;
  const float* w = Wu + (long)d * 64;               // Wu[0]
  float acc = s[d];
  #pragma unroll
  for (int t = 0; t < 64; t++) acc += row[t] * w[t];
  U[(long)b * 128 + d] = acc;
}

__global__ void k_u1(const float* __restrict__ Wu, const float* __restrict__ s,
                     float* __restrict__ U) {
  long idx = (long)blockIdx.x * 256 + threadIdx.x;  // B*64
  int b = (int)(idx >> 6), d = (int)(idx & 63);
  const float* row = U + (long)b * 128;             // u0
  const float* w = Wu + 4096 + (long)d * 64;        // Wu[1]
  float acc = s[64 + d];
  #pragma unroll
  for (int t = 0; t < 64; t++) acc += row[t] * w[t];
  U[(long)b * 128 + 64 + d] = acc;
}

// ---------------- rating = Ucat @ Qcat^T ----------------
__global__ void __launch_bounds__(32) k_rating(const float* __restrict__ U,
                                               const float* __restrict__ Q,
                                               float* __restrict__ out) {
  const int lane = threadIdx.x, h = lane >> 4, lm = lane & 15;
  const int m0 = blockIdx.x * 16, n0 = blockIdx.y * 16;
  v8f c = (v8f){0.f,0.f,0.f,0.f,0.f,0.f,0.f,0.f};
  #pragma unroll
  for (int s = 0; s < 32; s++) {
    int k0 = s * 4 + 2 * h;
    float2 av = *reinterpret_cast<const float2*>(&U[(long)(m0 + lm) * 128 + k0]);
    float2 bv = *reinterpret_cast<const float2*>(&Q[(long)(n0 + lm) * 128 + k0]);
    v2f a, b; a[0] = av.x; a[1] = av.y; b[0] = bv.x; b[1] = bv.y;
    c = wmma4(a, b, c);
  }
  #pragma unroll
  for (int v = 0; v < 8; v++)
    out[(long)(m0 + v + 8 * h) * NN + n0 + lm] = c[v];
}

extern "C" void kernel_launch(void* const* d_in, const int* in_sizes, int n_in,
                              void* d_out, int out_size, void* d_ws, size_t ws_size,
                              hipStream_t stream) {
  const int*   users = (const int*)d_in[0];
  const float* adj   = (const float*)d_in[3];
  const float* dist  = (const float*)d_in[4];
  const float* uemb  = (const float*)d_in[5];
  const float* pemb  = (const float*)d_in[6];
  const float* qemb  = (const float*)d_in[7];
  const float* Wg    = (const float*)d_in[8];
  const float* bg    = (const float*)d_in[9];
  const float* Wdec  = (const float*)d_in[10];
  const float* bdec  = (const float*)d_in[11];
  const float* pa    = (const float*)d_in[12];
  const float* pb    = (const float*)d_in[13];
  const float* pc    = (const float*)d_in[14];
  const float* Wu    = (const float*)d_in[15];
  const float* Wo    = (const float*)d_in[16];
  const float* bo    = (const float*)d_in[17];

  float* ws = (float*)d_ws;
  float* WpBuf = ws;                     // N*128
  float* WqBuf = WpBuf + (long)NN * 128; // N*128
  float* Pagg  = WqBuf + (long)NN * 128; // N*128
  float* Qagg  = Pagg  + (long)NN * 128; // N*128
  float* Dout  = Qagg  + (long)NN * 128; // N
  float* Din   = Dout  + NN;             // N
  float* Hbuf  = Din   + NN;             // N*64
  float* csP   = Hbuf  + (long)NN * 64;  // 128
  float* svec  = csP   + 128;            // 128
  float* Ucat  = svec  + 128;            // B*128

  float* outRating = (float*)d_out;                 // B*N
  float* outEhat   = (float*)d_out + (long)BB * NN; // N*N

  // 1) zero the atomically-accumulated region: Pagg, Qagg, Dout, Din (contiguous)
  long zeroN = (long)NN * 128 * 2 + 2 * NN;
  k_zero<<<4096, 256, 0, stream>>>(Pagg, zeroN);

  // 2) projections
  k_wpq<<<(2 * NN * 128) / 256, 256, 0, stream>>>(pemb, qemb, Wg, bg, WpBuf, WqBuf);

  // 3) aggregations (WMMA f32), fused degree sums
  k_agg_p<<<dim3(NN / 32, 8), 32, 0, stream>>>(adj, WpBuf, Pagg, Dout);
  k_agg_q<<<dim3(NN / 32, 8), 32, 0, stream>>>(adj, WqBuf, Qagg, Din);

  // 4) normalize + leaky relu
  k_epi<<<(NN * 128) / 256, 256, 0, stream>>>(Pagg, Qagg, Dout, Din);

  // 5) decoder H, then e_hat with fused fx (WMMA)
  k_h<<<(NN * 64) / 256, 256, 0, stream>>>(Pagg, Wdec, bdec, Hbuf);
  k_ehat<<<dim3(NN / 16, NN / 16), 32, 0, stream>>>(Hbuf, Qagg, dist, pa, pb, pc, outEhat);

  // 6) user tower
  k_colsum<<<128, 256, 0, stream>>>(Pagg, csP);
  k_svec<<<1, 128, 0, stream>>>(csP, Wo, bo, svec);
  k_u0<<<(BB * 64) / 256, 256, 0, stream>>>(users, uemb, Wu, svec, Ucat);
  k_u1<<<(BB * 64) / 256, 256, 0, stream>>>(Wu, svec, Ucat);

  // 7) rating (WMMA)
  k_rating<<<dim3(BB / 16, NN / 16), 32, 0, stream>>>(Ucat, Qagg, outRating);
}